// RNNBinaryClassifier_74062416053029
// MI455X (gfx1250) — compile-verified
//
#include <hip/hip_runtime.h>

// CDNA5 / gfx1250 fused 2-layer RNN scan, one wave32 per 16-row batch tile.
//   h1 = tanh(x_t @ Wih1^T + b1 + h1 @ Whh1^T)
//   h2 = tanh(h1  @ Wih2^T + b2 + h2 @ Whh2^T)
// Matmuls: V_WMMA_F32_16X16X4_F32, 64 per step, arranged as 6 independent
// accumulator chains (phase 1) + 4 short chains (phase 2) to cover XDL latency
// with a single wave per SIMD. x is streamed through an 8-slot LDS ring fed by
// global_load_async_to_lds_b128 (ASYNCcnt-throttled), which also performs the
// row-major -> A-fragment transpose. h-state D->A transpose bounces through a
// private LDS tile (row stride 36 f32: 16B-aligned rows, conflict-free for both
// the 2addr-store and b64-load patterns).

typedef __attribute__((ext_vector_type(2))) float v2f;
typedef __attribute__((ext_vector_type(8))) float v8f;

#define HDIM 32
#define LSTRIDE 36                 // LDS row stride in floats
#define LBUF (16 * LSTRIDE)        // 576 floats per 16x32 tile
#define RING_DEPTH 8
#define SLOT_BYTES (LBUF * 4)      // 2304 bytes per ring slot

__device__ __forceinline__ v8f wmma4(v2f a, v2f b, v8f c) {
  return __builtin_amdgcn_wmma_f32_16x16x4_f32(false, a, false, b, (short)0, c,
                                               false, false);
}

// tanh(s + bias) with bias pre-scaled: e = exp2(s*k + bias*k); 1 - 2/(e+1)
__device__ __forceinline__ float tanh_bias(float s, float bk) {
  float e = __builtin_amdgcn_exp2f(__builtin_fmaf(s, 2.8853900817779268f, bk));
  return 1.0f - 2.0f * __builtin_amdgcn_rcpf(e + 1.0f);
}

// One timestep tile (16 rows x 128B) global -> LDS, 4 async b128 ops.
// Lane L covers row L>>1, 16B segment (L&1); imm offsets walk the row, and the
// same imm applies to both global and LDS sides (rows are contiguous on both).
__device__ __forceinline__ void async_x_tile(unsigned lds_off,
                                             unsigned long long ga) {
  asm volatile(
      "global_load_async_to_lds_b128 %0, %1, off th:TH_LOAD_NT\n\t"
      "global_load_async_to_lds_b128 %0, %1, off offset:32 th:TH_LOAD_NT\n\t"
      "global_load_async_to_lds_b128 %0, %1, off offset:64 th:TH_LOAD_NT\n\t"
      "global_load_async_to_lds_b128 %0, %1, off offset:96 th:TH_LOAD_NT"
      :: "v"(lds_off), "v"(ga)
      : "memory");
}

__global__ void __launch_bounds__(32, 1)
rnn2_fused_scan(const float* __restrict__ x,
                const float* __restrict__ Wih1, const float* __restrict__ Whh1,
                const float* __restrict__ bih1, const float* __restrict__ bhh1,
                const float* __restrict__ Wih2, const float* __restrict__ Whh2,
                const float* __restrict__ bih2, const float* __restrict__ bhh2,
                const float* __restrict__ Wfc,  const float* __restrict__ bfc,
                float* __restrict__ out, int B, int T)
{
  const int lane  = threadIdx.x & 31;
  const int bbase = blockIdx.x * 16;        // one wave per 16 batch rows
  if (bbase >= B) return;                   // wave-uniform: EXEC stays all-1s

  const int half = lane >> 4;
  const int l15  = lane & 15;

  __shared__ __align__(16) float h1buf[LBUF];
  __shared__ __align__(16) float h2buf[LBUF];
  __shared__ __align__(16) float xring[RING_DEPTH][LBUF];

  // ---- Weight B-fragments, resident in VGPRs for the whole scan ----
  // Bm[k][n] = W[n][k]; frag(nt,c) elem j = W[16*nt + l15][4c + 2*half + j]
  v2f wih1f[2][8], whh1f[2][8], wih2f[2][8], whh2f[2][8];
  {
    const int ro = l15 * HDIM + 2 * half;
#pragma unroll
    for (int nt = 0; nt < 2; ++nt) {
#pragma unroll
      for (int c = 0; c < 8; ++c) {
        const int idx = (16 * nt) * HDIM + ro + 4 * c;
        wih1f[nt][c] = *(const v2f*)(Wih1 + idx);
        whh1f[nt][c] = *(const v2f*)(Whh1 + idx);
        wih2f[nt][c] = *(const v2f*)(Wih2 + idx);
        whh2f[nt][c] = *(const v2f*)(Whh2 + idx);
      }
    }
  }
  // Biases pre-scaled by 2*log2(e) so they fold into the tanh FMA.
  float bk1[2], bk2[2];
#pragma unroll
  for (int nt = 0; nt < 2; ++nt) {
    bk1[nt] = (bih1[16 * nt + l15] + bhh1[16 * nt + l15]) * 2.8853900817779268f;
    bk2[nt] = (bih2[16 * nt + l15] + bhh2[16 * nt + l15]) * 2.8853900817779268f;
  }

  // ---- h-state A-fragments (lane = batch row, VGPR j -> K = 4c + 2*half + j)
  v2f h1a[8], h2a[8];
  const v2f vz2 = {0.f, 0.f};
#pragma unroll
  for (int c = 0; c < 8; ++c) { h1a[c] = vz2; h2a[c] = vz2; }

  // ---- async x ring setup ----
  const unsigned xl0 =
      (unsigned)(size_t)(&xring[0][0]) +
      (unsigned)((lane >> 1) * (LSTRIDE * 4)) + (unsigned)((lane & 1) * 16);
  unsigned long long ga =
      (unsigned long long)(size_t)x +
      (unsigned long long)(bbase + (lane >> 1)) * (unsigned long long)T *
          (HDIM * 4) +
      (unsigned long long)((lane & 1) * 16);
  const int tLast = T - 1;
#pragma unroll
  for (int s = 0; s < RING_DEPTH - 1; ++s) {   // prefill slots 0..6 (cnt=28)
    async_x_tile(xl0 + (unsigned)s * SLOT_BYTES, ga);
    if (s < tLast) ga += HDIM * 4;             // clamp at last timestep
  }

  const int stb = half * (8 * LSTRIDE) + l15;  // D-layout store base (floats)
  const int ldb = l15 * LSTRIDE + 2 * half;    // A-layout load  base (floats)
  const v8f z8 = {0.f, 0.f, 0.f, 0.f, 0.f, 0.f, 0.f, 0.f};

  for (int t = 0; t < T; ++t) {
    // Oldest in-flight group (slot t) must be complete: 7 groups -> wait <= 24.
    asm volatile("s_wait_asynccnt 24" ::: "memory");

    // x A-fragments for step t straight from the LDS ring (ds_load_b64 x8).
    const float* xslot = &xring[t & (RING_DEPTH - 1)][0];
    v2f xa[8];
#pragma unroll
    for (int c = 0; c < 8; ++c)
      xa[c] = *(const v2f*)(xslot + ldb + 4 * c);

    // Refill: slot for timestep t+7 lands in ring slot (t-1)&7 (dead slot).
    // Address is clamped near the end -> always 4 ops, invariant count stays 28.
    async_x_tile(xl0 + (unsigned)((t + RING_DEPTH - 1) & (RING_DEPTH - 1)) *
                           SLOT_BYTES,
                 ga);
    if (t + RING_DEPTH - 1 < tLast) ga += HDIM * 4;

    // -------- phase 1: six independent WMMA chains (48 WMMAs) --------
    v8f ax1[2], ah1[2], ah2[2];
#pragma unroll
    for (int nt = 0; nt < 2; ++nt) { ax1[nt] = z8; ah1[nt] = z8; ah2[nt] = z8; }
#pragma unroll
    for (int c = 0; c < 8; ++c) {
#pragma unroll
      for (int nt = 0; nt < 2; ++nt) {
        ax1[nt] = wmma4(xa[c],  wih1f[nt][c], ax1[nt]);   // x_t  @ Wih1^T
        ah1[nt] = wmma4(h1a[c], whh1f[nt][c], ah1[nt]);   // h1   @ Whh1^T
        ah2[nt] = wmma4(h2a[c], whh2f[nt][c], ah2[nt]);   // h2   @ Whh2^T
      }
    }
    // layer-1 tanh in D-layout -> LDS -> rebuild A-fragments
#pragma unroll
    for (int nt = 0; nt < 2; ++nt)
#pragma unroll
      for (int r = 0; r < 8; ++r)
        h1buf[stb + r * LSTRIDE + nt * 16] =
            tanh_bias(ax1[nt][r] + ah1[nt][r], bk1[nt]);
#pragma unroll
    for (int c = 0; c < 8; ++c)
      h1a[c] = *(const v2f*)(h1buf + ldb + 4 * c);

    // -------- phase 2: layer-2 input term, 4 chains of 4 (16 WMMAs) --------
    v8f ax2a[2], ax2b[2];
#pragma unroll
    for (int nt = 0; nt < 2; ++nt) { ax2a[nt] = z8; ax2b[nt] = z8; }
#pragma unroll
    for (int c = 0; c < 4; ++c) {
#pragma unroll
      for (int nt = 0; nt < 2; ++nt) {
        ax2a[nt] = wmma4(h1a[c],     wih2f[nt][c],     ax2a[nt]);
        ax2b[nt] = wmma4(h1a[c + 4], wih2f[nt][c + 4], ax2b[nt]);
      }
    }
#pragma unroll
    for (int nt = 0; nt < 2; ++nt)
#pragma unroll
      for (int r = 0; r < 8; ++r)
        h2buf[stb + r * LSTRIDE + nt * 16] =
            tanh_bias(ax2a[nt][r] + ax2b[nt][r] + ah2[nt][r], bk2[nt]);
#pragma unroll
    for (int c = 0; c < 8; ++c)
      h2a[c] = *(const v2f*)(h2buf + ldb + 4 * c);
  }

  // ---- final FC: out[b] = h2[b,:] . Wfc + bfc ; h2 sits in LDS (b x 36) ----
  if (lane < 16) {
    float sum = 0.f;
#pragma unroll
    for (int h = 0; h < HDIM; ++h)
      sum += h2buf[lane * LSTRIDE + h] * Wfc[h];
    out[bbase + lane] = sum + bfc[0];
  }
}

extern "C" void kernel_launch(void* const* d_in, const int* in_sizes, int n_in,
                              void* d_out, int out_size, void* d_ws, size_t ws_size,
                              hipStream_t stream) {
  const float* x    = (const float*)d_in[0];
  const float* Wih1 = (const float*)d_in[1];
  const float* Whh1 = (const float*)d_in[2];
  const float* bih1 = (const float*)d_in[3];
  const float* bhh1 = (const float*)d_in[4];
  const float* Wih2 = (const float*)d_in[5];
  const float* Whh2 = (const float*)d_in[6];
  const float* bih2 = (const float*)d_in[7];
  const float* bhh2 = (const float*)d_in[8];
  const float* Wfc  = (const float*)d_in[9];
  const float* bfc  = (const float*)d_in[10];
  float* out = (float*)d_out;

  const int B = out_size;                  // one scalar per batch row
  const int T = in_sizes[0] / (B * HDIM);  // x is (B, T, H)
  const int tiles = (B + 15) / 16;         // one wave32 (one block) per tile

  rnn2_fused_scan<<<tiles, 32, 0, stream>>>(
      x, Wih1, Whh1, bih1, bhh1, Wih2, Whh2, bih2, bhh2, Wfc, bfc, out, B, T);
}